// ROIAlignModule_33423435498486
// MI455X (gfx1250) — compile-verified
//
#include <hip/hip_runtime.h>
#include <hip/hip_bf16.h>
#include <math.h>

typedef float v2f __attribute__((ext_vector_type(2)));
typedef float v8f __attribute__((ext_vector_type(8)));

#define EPS 1e-5f

__device__ __forceinline__ float hswish(float v) {
    return v * fminf(fmaxf(v + 3.0f, 0.0f), 6.0f) * (1.0f / 6.0f);
}

// ---------------------------------------------------------------------------
// Fused 1x1x1 conv (GEMM) + BatchNorm + hswish using V_WMMA_F32_16X16X4_F32.
// Out[n][m][s] = hswish(bn(sum_k W[m][k] * In[n][k][s])),  M = 128 fixed.
// Block: 256 threads = 8 waves; wave w owns rows [16w, 16w+16).
// Each block handles one (sample n, 16-wide column tile s_base).
// LDS: activation tile staged transposed xs[s][k], row pitch 260 floats
//      (pitch%64 == 4 -> 16 distinct banks for the 16 s-values per ds read).
// ---------------------------------------------------------------------------
__global__ __launch_bounds__(256) void gemm_bn_hswish_wmma(
    const float* __restrict__ W,    // [128 x K] row-major
    const float* __restrict__ In,   // [nsamp][K][S]
    float* __restrict__ Out,        // [nsamp][128][S]
    const float* __restrict__ gg, const float* __restrict__ bb,
    const float* __restrict__ mm, const float* __restrict__ vv,
    int K, int S)
{
    __shared__ float xs[16][260];

    const int s_tiles = S >> 4;
    const int n       = blockIdx.x / s_tiles;
    const int s_base  = (blockIdx.x % s_tiles) << 4;
    const int t       = threadIdx.x;

    // Cooperative stage: In[n][k][s_base..s_base+15] -> xs[s][k] (transposed)
    for (int k = t; k < K; k += 256) {
        const float* src = In + ((size_t)n * K + k) * S + s_base;
        #pragma unroll
        for (int s = 0; s < 16; ++s) xs[s][k] = src[s];
    }
    __syncthreads();

    const int wave   = t >> 5;          // 0..7 -> M tile
    const int lane   = t & 31;
    const int half   = lane >> 4;       // 0: K+0/K+1,  1: K+2/K+3
    const int ml     = lane & 15;       // A row within tile / B column
    const int m_base = wave << 4;

    v8f acc = {};
    const float* wrow = W + (size_t)(m_base + ml) * K;

    for (int kk = 0; kk < K; kk += 4) {
        const int ka = kk + (half << 1);
        v2f a = *(const v2f*)(wrow + ka);           // A: W[m][ka], W[m][ka+1]
        v2f b = *(const v2f*)(&xs[ml][ka]);         // B: In[ka][s], In[ka+1][s]
        acc = __builtin_amdgcn_wmma_f32_16x16x4_f32(
            false, a, false, b, (short)0, acc, false, false);
    }

    // Epilogue: C/D layout -> VGPR i holds row m = m_base + i + 8*half, col ml.
    #pragma unroll
    for (int i = 0; i < 8; ++i) {
        const int m   = m_base + i + (half << 3);
        const float sc = gg[m] * rsqrtf(vv[m] + EPS);
        const float sh = bb[m] - mm[m] * sc;
        const float v  = acc[i] * sc + sh;
        Out[((size_t)n * 128 + m) * S + s_base + ml] = hswish(v);
    }
}

// ---------------------------------------------------------------------------
// Depthwise 3x3x3 conv, stride 2, pad 1 (+BN2+hswish).
// In  [64][128][8][16][16] -> Out [64][128][4][8][8]
// ---------------------------------------------------------------------------
__global__ __launch_bounds__(256) void dwconv_bn_hswish(
    const float* __restrict__ In, const float* __restrict__ Wd,  // [128][27]
    float* __restrict__ Out,
    const float* __restrict__ gg, const float* __restrict__ bb,
    const float* __restrict__ mm, const float* __restrict__ vv,
    int total)
{
    const int idx = blockIdx.x * 256 + threadIdx.x;
    if (idx >= total) return;
    const int w  = idx & 7;
    const int h  = (idx >> 3) & 7;
    const int tt = (idx >> 6) & 3;
    const int c  = (idx >> 8) & 127;
    const int n  = idx >> 15;

    const float* wp = Wd + c * 27;
    const float* ip = In + ((size_t)n * 128 + c) * 2048;

    float acc = 0.0f;
    #pragma unroll
    for (int kt = 0; kt < 3; ++kt) {
        const int ti = tt * 2 - 1 + kt;
        if (ti < 0 || ti > 7) continue;
        #pragma unroll
        for (int kh = 0; kh < 3; ++kh) {
            const int hi = h * 2 - 1 + kh;
            if (hi < 0 || hi > 15) continue;
            #pragma unroll
            for (int kw = 0; kw < 3; ++kw) {
                const int wi = w * 2 - 1 + kw;
                if (wi < 0 || wi > 15) continue;
                acc += wp[(kt * 3 + kh) * 3 + kw] * ip[ti * 256 + hi * 16 + wi];
            }
        }
    }
    const float sc = gg[c] * rsqrtf(vv[c] + EPS);
    Out[idx] = hswish(acc * sc + (bb[c] - mm[c] * sc));
}

// ---------------------------------------------------------------------------
// Global mean over S=256, 6-channel logits (w3 dot), BN4, sigmoid, ROI params.
// roi[n][6] = {start_x, start_y, start_t, end_x, end_y, end_t}
// ---------------------------------------------------------------------------
__global__ __launch_bounds__(128) void pool_logits_roi(
    const float* __restrict__ h3,   // [64][128][256]
    const float* __restrict__ w3,   // [6][128]
    const float* __restrict__ g4, const float* __restrict__ b4,
    const float* __restrict__ m4, const float* __restrict__ v4,
    float* __restrict__ roi)        // [64][6]
{
    __shared__ float pooled[128];
    __shared__ float lim[6];
    const int n = blockIdx.x;
    const int c = threadIdx.x;

    const float* p = h3 + ((size_t)n * 128 + c) * 256;
    float s = 0.0f;
    for (int i = 0; i < 256; ++i) s += p[i];
    pooled[c] = s * (1.0f / 256.0f);
    __syncthreads();

    if (c < 6) {
        float z = 0.0f;
        for (int k = 0; k < 128; ++k) z += w3[c * 128 + k] * pooled[k];
        const float sc = g4[c] * rsqrtf(v4[c] + EPS);
        z = z * sc + (b4[c] - m4[c] * sc);
        lim[c] = 1.0f / (1.0f + __expf(-z));
    }
    __syncthreads();
    if (c < 6) {
        const float scl = (c % 3 == 0) ? 16.0f : ((c % 3 == 1) ? 16.0f : 8.0f);
        roi[n * 6 + c] = (c < 3) ? 0.5f * lim[c] * scl
                                 : (1.0f - 0.5f * lim[c]) * scl;
    }
}

// ---------------------------------------------------------------------------
// 3D ROI align on original x: out [64][256][4][7][7], RS=2 (8 samples/bin).
// ---------------------------------------------------------------------------
__global__ __launch_bounds__(256) void roi_align3d(
    const float* __restrict__ x, const float* __restrict__ roi,
    float* __restrict__ out, int total)
{
    const int idx = blockIdx.x * 256 + threadIdx.x;
    if (idx >= total) return;
    const int pw = idx % 7;
    const int ph = (idx / 7) % 7;
    const int pt = (idx / 49) % 4;
    const int c  = (idx / 196) & 255;
    const int n  = idx / 50176;           // 196*256

    const float* r = roi + n * 6;
    const float sx = r[0], sy = r[1], st = r[2];
    const float bx = fmaxf(r[3] - sx, 1.0f) * (1.0f / 7.0f);
    const float by = fmaxf(r[4] - sy, 1.0f) * (1.0f / 7.0f);
    const float bt = fmaxf(r[5] - st, 1.0f) * (1.0f / 4.0f);

    const float* f = x + ((size_t)n * 256 + c) * 2048;

    float acc = 0.0f;
    #pragma unroll
    for (int i = 0; i < 2; ++i) {
        float ct = st + ((float)(pt * 2 + i) + 0.5f) * 0.5f * bt;
        ct = fminf(fmaxf(ct, 0.0f), 7.0f);
        const int t0 = (int)floorf(ct);
        const int t1 = min(t0 + 1, 7);
        const float wt = ct - (float)t0;
        #pragma unroll
        for (int j = 0; j < 2; ++j) {
            float cy = sy + ((float)(ph * 2 + j) + 0.5f) * 0.5f * by;
            cy = fminf(fmaxf(cy, 0.0f), 15.0f);
            const int y0 = (int)floorf(cy);
            const int y1 = min(y0 + 1, 15);
            const float wy = cy - (float)y0;
            #pragma unroll
            for (int k = 0; k < 2; ++k) {
                float cx = sx + ((float)(pw * 2 + k) + 0.5f) * 0.5f * bx;
                cx = fminf(fmaxf(cx, 0.0f), 15.0f);
                const int x0 = (int)floorf(cx);
                const int x1 = min(x0 + 1, 15);
                const float wx = cx - (float)x0;

                const float* f0 = f + t0 * 256;
                const float* f1 = f + t1 * 256;
                const float v0 =
                    (f0[y0 * 16 + x0] * (1.0f - wx) + f0[y0 * 16 + x1] * wx) * (1.0f - wy) +
                    (f0[y1 * 16 + x0] * (1.0f - wx) + f0[y1 * 16 + x1] * wx) * wy;
                const float v1 =
                    (f1[y0 * 16 + x0] * (1.0f - wx) + f1[y0 * 16 + x1] * wx) * (1.0f - wy) +
                    (f1[y1 * 16 + x0] * (1.0f - wx) + f1[y1 * 16 + x1] * wx) * wy;
                acc += v0 * (1.0f - wt) + v1 * wt;
            }
        }
    }
    out[idx] = acc * 0.125f;
}

// ---------------------------------------------------------------------------
extern "C" void kernel_launch(void* const* d_in, const int* in_sizes, int n_in,
                              void* d_out, int out_size, void* d_ws, size_t ws_size,
                              hipStream_t stream) {
    const float* x   = (const float*)d_in[0];
    const float* w1  = (const float*)d_in[1];
    const float* g1  = (const float*)d_in[2];
    const float* b1  = (const float*)d_in[3];
    const float* m1  = (const float*)d_in[4];
    const float* v1  = (const float*)d_in[5];
    const float* wdw = (const float*)d_in[6];
    const float* g2  = (const float*)d_in[7];
    const float* b2  = (const float*)d_in[8];
    const float* m2  = (const float*)d_in[9];
    const float* v2  = (const float*)d_in[10];
    const float* w2  = (const float*)d_in[11];
    const float* g3  = (const float*)d_in[12];
    const float* b3  = (const float*)d_in[13];
    const float* m3  = (const float*)d_in[14];
    const float* v3  = (const float*)d_in[15];
    const float* w3  = (const float*)d_in[16];
    const float* g4  = (const float*)d_in[17];
    const float* b4  = (const float*)d_in[18];
    const float* m4  = (const float*)d_in[19];
    const float* v4  = (const float*)d_in[20];

    float* h1  = (float*)d_ws;                       // [64][128][2048]
    float* h2  = h1 + (size_t)64 * 128 * 2048;       // [64][128][256]
    float* h3  = h2 + (size_t)64 * 128 * 256;        // [64][128][256]
    float* roi = h3 + (size_t)64 * 128 * 256;        // [64][6]

    // conv1 (256->128 over S=2048) + bn1 + hswish
    gemm_bn_hswish_wmma<<<64 * 128, 256, 0, stream>>>(
        w1, x, h1, g1, b1, m1, v1, 256, 2048);

    // depthwise 3x3x3 s2 p1 + bn2 + hswish
    dwconv_bn_hswish<<<(64 * 128 * 256) / 256, 256, 0, stream>>>(
        h1, wdw, h2, g2, b2, m2, v2, 64 * 128 * 256);

    // conv2 (128->128 over S=256) + bn3 + hswish
    gemm_bn_hswish_wmma<<<64 * 16, 256, 0, stream>>>(
        w2, h2, h3, g3, b3, m3, v3, 128, 256);

    // global mean + logits + sigmoid -> ROI start/end
    pool_logits_roi<<<64, 128, 0, stream>>>(h3, w3, g4, b4, m4, v4, roi);

    // trilinear ROI align over original x
    const int total = 64 * 256 * 4 * 7 * 7;
    roi_align3d<<<(total + 255) / 256, 256, 0, stream>>>(x, roi, (float*)d_out, total);
}